// BSLoss_tb_new_52286931861540
// MI455X (gfx1250) — compile-verified
//
#include <hip/hip_runtime.h>
#include <hip/hip_bf16.h>

typedef float v2f __attribute__((ext_vector_type(2)));
typedef float v8f __attribute__((ext_vector_type(8)));

#define KDIM 16
// ws layout in 4-byte words
#define ACC_OFF   0                  // 3 * 16 floats: [npos, negcnt, losspos, tclpos, tclneg, rx, ry, sumgt, ...]
#define SEL_OFF   48                 // 3 * 8 uints:  [nneg, b1, b2, b3, need, cntgt, ...]
#define H1_OFF    72                 // 3 * 1024
#define H2_OFF    (72 + 3072)        // 3 * 2048
#define H3_OFF    (72 + 3072 + 6144) // 3 * 1024
#define WS_WORDS  (72 + 3072 + 6144 + 3072)

static __device__ __forceinline__ float ce2(float z0, float z1, bool t1) {
  float m = fmaxf(z0, z1);
  float lse = m + __logf(__expf(z0 - m) + __expf(z1 - m));
  return lse - (t1 ? z1 : z0);
}

static __device__ __forceinline__ float sl1(float a, float b) {
  float d = fabsf(a - b);
  return (d < 1.0f) ? (0.5f * d * d) : (d - 0.5f);
}

// ---------------- main streaming pass: per-pixel CE + masked sums ----------------
__global__ void level_main(const float* __restrict__ cls,
                           const float* __restrict__ reg,
                           const float* __restrict__ gt,
                           float* __restrict__ acc,
                           int HW, int npix)
{
  float q0 = 0.f, q1 = 0.f, q2 = 0.f, q3 = 0.f, q4 = 0.f, q5 = 0.f, q6 = 0.f;
  int tid = blockIdx.x * blockDim.x + threadIdx.x;
  int p0 = tid * 4;
  if (p0 < npix) {
    int b  = p0 / HW;
    int hw = p0 - b * HW;
    const float4 tr  = *(const float4*)(gt  + (size_t)(b * 35 + 0) * HW + hw);
    const float4 tcl = *(const float4*)(gt  + (size_t)(b * 35 + 1) * HW + hw);
    const float4 trn = *(const float4*)(gt  + (size_t)(b * 35 + 2) * HW + hw);
    const float4 z0  = *(const float4*)(cls + (size_t)(b * 4  + 0) * HW + hw);
    const float4 z1  = *(const float4*)(cls + (size_t)(b * 4  + 1) * HW + hw);
    const float4 z2  = *(const float4*)(cls + (size_t)(b * 4  + 2) * HW + hw);
    const float4 z3  = *(const float4*)(cls + (size_t)(b * 4  + 3) * HW + hw);
    const float* xm = gt  + (size_t)(b * 35 + 3) * HW + hw;
    const float* xp = reg + (size_t)(b * 32 + 0) * HW + hw;
    float4 sx = make_float4(0.f, 0.f, 0.f, 0.f);
    float4 sy = make_float4(0.f, 0.f, 0.f, 0.f);
#pragma unroll
    for (int j = 0; j < KDIM; ++j) {
      float4 a  = *(const float4*)(xm + (size_t)j * HW);
      float4 p  = *(const float4*)(xp + (size_t)j * HW);
      float4 ay = *(const float4*)(xm + (size_t)(KDIM + j) * HW);
      float4 py = *(const float4*)(xp + (size_t)(KDIM + j) * HW);
      sx.x += sl1(a.x, p.x);  sx.y += sl1(a.y, p.y);
      sx.z += sl1(a.z, p.z);  sx.w += sl1(a.w, p.w);
      sy.x += sl1(ay.x, py.x); sy.y += sl1(ay.y, py.y);
      sy.z += sl1(ay.z, py.z); sy.w += sl1(ay.w, py.w);
    }
#define COMPONENT(c)                                                     \
    {                                                                    \
      float trv = tr.c, tclv = tcl.c, tnv = trn.c;                       \
      bool posf = (trv > 0.5f) && (tnv > 0.5f);                          \
      bool negf = (trv < 0.5f) && (tnv > 0.5f);                          \
      float cet = ce2(z0.c, z1.c, trv > 0.5f);                           \
      float cec = ce2(z2.c, z3.c, tclv > 0.5f);                          \
      if (posf) {                                                        \
        q0 += 1.f; q2 += cet; q3 += cec;                                 \
        float w = 0.5f * (trv + tclv);                                   \
        q5 += w * sx.c; q6 += w * sy.c;                                  \
      } else {                                                           \
        q4 += cec;                                                       \
      }                                                                  \
      if (negf) q1 += 1.f;                                               \
    }
    COMPONENT(x) COMPONENT(y) COMPONENT(z) COMPONENT(w)
#undef COMPONENT
  }

  // stage 1: wave32 shuffle reduction of the 7 partials
  float vals[7] = {q0, q1, q2, q3, q4, q5, q6};
#pragma unroll
  for (int q = 0; q < 7; ++q) {
    float v = vals[q];
#pragma unroll
    for (int o = 16; o > 0; o >>= 1) v += __shfl_xor(v, o, 32);
    vals[q] = v;
  }

  // stage 2: 8 wave-partials x 7 quantities reduced by one V_WMMA_F32_16X16X4_F32
  __shared__ float part[16][8];
  if (threadIdx.x < 128) ((float*)part)[threadIdx.x] = 0.f;
  __syncthreads();
  int lane = threadIdx.x & 31;
  int wv   = threadIdx.x >> 5;
  if (lane == 0) {
#pragma unroll
    for (int q = 0; q < 7; ++q) part[q][wv] = vals[q];
  }
  __syncthreads();
  if (threadIdx.x < 32) {            // wave 0, EXEC all ones
    int l  = threadIdx.x;
    int m  = l & 15;
    int kh = l >> 4;
    // A[m][k] = part[m][k] + part[m][k+4]; lane layout: vgpr j, lane l -> K = 2*(l>>4)+j
    v2f a;
    a.x = part[m][2 * kh + 0] + part[m][2 * kh + 4];
    a.y = part[m][2 * kh + 1] + part[m][2 * kh + 5];
    v2f bb; bb.x = 1.0f; bb.y = 1.0f;      // B = ones(4x16)
    v8f c = {};
    v8f d = __builtin_amdgcn_wmma_f32_16x16x4_f32(false, a, false, bb,
                                                  (short)0, c, false, false);
    if (l == 0) {
      atomicAdd(&acc[0], d[0]); atomicAdd(&acc[1], d[1]);
      atomicAdd(&acc[2], d[2]); atomicAdd(&acc[3], d[3]);
      atomicAdd(&acc[4], d[4]); atomicAdd(&acc[5], d[5]);
      atomicAdd(&acc[6], d[6]);
    }
  }
}

// ---------------- radix-select histogram passes over negative-pixel CE ----------------
__global__ void hist_pass(const float* __restrict__ cls,
                          const float* __restrict__ gt,
                          const unsigned* __restrict__ sel,
                          unsigned* __restrict__ hist,
                          int HW, int npix, int phase)
{
  __shared__ unsigned hsh[1024];
  if (phase == 1) {
    for (int i = threadIdx.x; i < 1024; i += blockDim.x) hsh[i] = 0u;
    __syncthreads();
  }
  int p = blockIdx.x * blockDim.x + threadIdx.x;
  if (p < npix) {
    int b  = p / HW;
    int hw = p - b * HW;
    float trv = gt[(size_t)(b * 35 + 0) * HW + hw];
    float tnv = gt[(size_t)(b * 35 + 2) * HW + hw];
    if (trv < 0.5f && tnv > 0.5f) {
      float z0v = cls[(size_t)(b * 4 + 0) * HW + hw];
      float z1v = cls[(size_t)(b * 4 + 1) * HW + hw];
      float ce  = fmaxf(ce2(z0v, z1v, false), 0.f);  // target==0 for neg pixels
      unsigned bits = __float_as_uint(ce);
      if (phase == 1) {
        atomicAdd(&hsh[bits >> 21], 1u);
      } else if (phase == 2) {
        if ((bits >> 21) == sel[1]) atomicAdd(&hist[(bits >> 10) & 0x7FFu], 1u);
      } else {
        if ((bits >> 10) == ((sel[1] << 11) | sel[2])) atomicAdd(&hist[bits & 0x3FFu], 1u);
      }
    }
  }
  if (phase == 1) {
    __syncthreads();
    for (int i = threadIdx.x; i < 1024; i += blockDim.x)
      if (hsh[i]) atomicAdd(&hist[i], hsh[i]);
  }
}

__global__ void scan_pass(const unsigned* __restrict__ hist,
                          const float* __restrict__ acc,
                          unsigned* __restrict__ sel,
                          int nbins, int phase)
{
  __shared__ unsigned hsh[2048];
  __shared__ unsigned csum[256];
  int t = threadIdx.x;
  for (int i = t; i < nbins; i += 256) hsh[i] = hist[i];
  __syncthreads();
  int per = nbins / 256;
  unsigned s = 0;
  for (int j = 0; j < per; ++j) s += hsh[t * per + j];
  csum[t] = s;
  __syncthreads();
  if (t == 0) {
    unsigned need;
    if (phase == 1) {
      float npos = acc[0], negc = acc[1];
      unsigned nneg;
      if (npos > 0.f) {
        unsigned a1 = (unsigned)negc;
        unsigned a2 = (unsigned)(3.0f * npos);   // floor(ratio*n_pos), exact
        nneg = (a1 < a2) ? a1 : a2;
      } else {
        nneg = 100u;
      }
      sel[0] = nneg;
      need = nneg;
    } else {
      need = sel[4];
    }
    unsigned run = 0;
    int ci = -1;
    for (int i = 255; i >= 0; --i) {
      if (need > 0u && run + csum[i] >= need) { ci = i; break; }
      run += csum[i];
    }
    int bsel = 0;
    unsigned above;
    if (ci >= 0) {
      above = run;
      for (int j = per - 1; j >= 0; --j) {
        int b = ci * per + j;
        if (above + hsh[b] >= need) { bsel = b; break; }
        above += hsh[b];
      }
    } else {
      bsel = 0;
      above = run - hsh[0];   // take-all fallback: threshold collapses to bin 0
    }
    sel[phase] = (unsigned)bsel;
    sel[4] = need - above;
  }
}

// ---------------- final pass: sum/count CE strictly above exact threshold ----------------
__global__ void final_sum(const float* __restrict__ cls,
                          const float* __restrict__ gt,
                          unsigned* __restrict__ sel,
                          float* __restrict__ acc,
                          int HW, int npix)
{
  __shared__ float ssum;
  __shared__ unsigned scnt;
  if (threadIdx.x == 0) { ssum = 0.f; scnt = 0u; }
  __syncthreads();
  int p = blockIdx.x * blockDim.x + threadIdx.x;
  if (p < npix) {
    int b  = p / HW;
    int hw = p - b * HW;
    float trv = gt[(size_t)(b * 35 + 0) * HW + hw];
    float tnv = gt[(size_t)(b * 35 + 2) * HW + hw];
    if (trv < 0.5f && tnv > 0.5f) {
      float z0v = cls[(size_t)(b * 4 + 0) * HW + hw];
      float z1v = cls[(size_t)(b * 4 + 1) * HW + hw];
      float ce  = fmaxf(ce2(z0v, z1v, false), 0.f);
      unsigned T = (sel[1] << 21) | (sel[2] << 10) | sel[3];
      float thr = __uint_as_float(T);
      if (ce > thr) { atomicAdd(&ssum, ce); atomicAdd(&scnt, 1u); }
    }
  }
  __syncthreads();
  if (threadIdx.x == 0 && scnt) {
    atomicAdd(&acc[7], ssum);
    atomicAdd(&sel[5], scnt);
  }
}

// ---------------- combine the 3 levels into the 4 outputs ----------------
__global__ void combine_k(const float* __restrict__ ws_acc,
                          const unsigned* __restrict__ ws_sel,
                          float* __restrict__ out)
{
  if (threadIdx.x != 0 || blockIdx.x != 0) return;
  const int Ns[3] = {409600, 102400, 25600};
  float o0 = 0.f, o1 = 0.f, o2 = 0.f, o3 = 0.f;
  for (int l = 0; l < 3; ++l) {
    const float* acc    = ws_acc + l * 16;
    const unsigned* sel = ws_sel + l * 8;
    float npos = acc[0];
    float lpos = acc[2], tclp = acc[3], tcln = acc[4], rx = acc[5], ry = acc[6];
    float sumgt = acc[7];
    unsigned nneg  = sel[0];
    unsigned cntgt = sel[5];
    float lneg = 0.f;
    if (nneg > 0u) {
      unsigned T = (sel[1] << 21) | (sel[2] << 10) | sel[3];
      float thr = __uint_as_float(T);
      lneg = sumgt + ((float)nneg - (float)cntgt) * thr;  // exact ties at threshold
    }
    float ltr = (lpos + lneg) / (npos + (float)nneg);
    bool has = npos > 0.f;
    float nnegm = (float)Ns[l] - npos;
    float ltcl = has ? (tclp / fmaxf(npos, 1.f) + 0.5f * tcln / fmaxf(nnegm, 1.f)) : 0.f;
    float denom = fmaxf(npos, 1.f) * (float)KDIM;
    float lrx = has ? rx / denom : 0.f;
    float lry = has ? ry / denom : 0.f;
    o0 += ltr; o1 += ltcl; o2 += lrx; o3 += lry;
  }
  out[0] = o0; out[1] = o1; out[2] = o2; out[3] = o3;
}

__global__ void ws_init(unsigned* __restrict__ ws)
{
  int i = blockIdx.x * blockDim.x + threadIdx.x;
  if (i < WS_WORDS) ws[i] = 0u;
}

extern "C" void kernel_launch(void* const* d_in, const int* in_sizes, int n_in,
                              void* d_out, int out_size, void* d_ws, size_t ws_size,
                              hipStream_t stream)
{
  unsigned* ws = (unsigned*)d_ws;
  float* accf  = (float*)d_ws;
  const int HWs[3] = {25600, 6400, 1600};

  ws_init<<<(WS_WORDS + 255) / 256, 256, 0, stream>>>(ws);

  for (int l = 0; l < 3; ++l) {
    const float* cls = (const float*)d_in[3 * l + 0];
    const float* reg = (const float*)d_in[3 * l + 1];
    const float* gt  = (const float*)d_in[3 * l + 2];
    int HW = HWs[l];
    int npix = 16 * HW;
    float*    acc = accf + ACC_OFF + l * 16;
    unsigned* sel = ws + SEL_OFF + l * 8;
    unsigned* h1  = ws + H1_OFF + l * 1024;
    unsigned* h2  = ws + H2_OFF + l * 2048;
    unsigned* h3  = ws + H3_OFF + l * 1024;

    int nthr4 = npix / 4;
    level_main<<<(nthr4 + 255) / 256, 256, 0, stream>>>(cls, reg, gt, acc, HW, npix);

    int nb = (npix + 255) / 256;
    hist_pass<<<nb, 256, 0, stream>>>(cls, gt, sel, h1, HW, npix, 1);
    scan_pass<<<1, 256, 0, stream>>>(h1, acc, sel, 1024, 1);
    hist_pass<<<nb, 256, 0, stream>>>(cls, gt, sel, h2, HW, npix, 2);
    scan_pass<<<1, 256, 0, stream>>>(h2, acc, sel, 2048, 2);
    hist_pass<<<nb, 256, 0, stream>>>(cls, gt, sel, h3, HW, npix, 3);
    scan_pass<<<1, 256, 0, stream>>>(h3, acc, sel, 1024, 3);
    final_sum<<<nb, 256, 0, stream>>>(cls, gt, sel, acc, HW, npix);
  }

  combine_k<<<1, 1, 0, stream>>>(accf + ACC_OFF, ws + SEL_OFF, (float*)d_out);
}